// ManyBodyPotential_85143431675992
// MI455X (gfx1250) — compile-verified
//
#include <hip/hip_runtime.h>

typedef __attribute__((ext_vector_type(2))) float v2f;
typedef __attribute__((ext_vector_type(8))) float v8f;

#define MBP_EPS 1e-12f

__global__ void mbp_zero_acc(double* acc) { acc[0] = 0.0; }

__global__ __launch_bounds__(256) void mbp_kernel(
    const float* __restrict__ pos,       // [num_atoms*3]
    const int*   __restrict__ edge_tgt,  // target row of edge_index, [num_atoms*16]
    const int*   __restrict__ elem,      // [num_atoms]
    const float* __restrict__ k_angle,   // [6]
    const float* __restrict__ cos0,      // [6]
    const int*   __restrict__ plt,       // [9] row-major 3x3
    int num_atoms,
    double* __restrict__ acc)
{
    // Fused lookup table: tbl[eA*3+eB] = (k_angle[p], cos0[p]), p = plt[eA][eB]
    __shared__ float2 tbl[9];
    if (threadIdx.x < 9) {
        int p = plt[threadIdx.x];
        tbl[threadIdx.x] = make_float2(k_angle[p], cos0[p]);
    }
    __syncthreads();

    const int lane          = threadIdx.x & 31;
    const int wave_in_blk   = threadIdx.x >> 5;
    const int waves_per_blk = blockDim.x >> 5;
    const int wave_id       = blockIdx.x * waves_per_blk + wave_in_blk;
    const int n_waves       = gridDim.x * waves_per_blk;
    const int row           = lane & 15;         // this lane's neighbor row
    const bool lo           = lane < 16;
    const int mbase         = lo ? 0 : 8;
    const char* tb          = (const char*)tbl;

    double dsum = 0.0;

    // One atom per wave per iteration; loop bound is wave-uniform -> EXEC all 1s at WMMA.
    for (int atom = wave_id; atom < num_atoms; atom += n_waves) {
        // Gather this row's neighbor displacement (both wave halves duplicate rows 0..15).
        int   t   = edge_tgt[atom * 16 + row];
        float pix = pos[atom * 3 + 0];
        float piy = pos[atom * 3 + 1];
        float piz = pos[atom * 3 + 2];
        float rx  = pos[t * 3 + 0] - pix;
        float ry  = pos[t * 3 + 1] - piy;
        float rz  = pos[t * 3 + 2] - piz;
        float ef  = (float)elem[t];              // element index as float (0,1,2 exact)
        float n   = __builtin_amdgcn_sqrtf(rx * rx + ry * ry + rz * rz); // v_sqrt_f32

        // A (16x4 f32) layout: lanes 0-15 hold K=0,1; lanes 16-31 hold K=2,3.
        // B = A^T (4x16) has the mirrored striping -> identical register images.

        // (1) Gram matrix of displacements: D[M][N] = dot(r_M, r_N)
        v2f a_dot;
        a_dot.x = lo ? rx : rz;
        a_dot.y = lo ? ry : 0.0f;
        v8f c_dot = {};
        c_dot = __builtin_amdgcn_wmma_f32_16x16x4_f32(
                    false, a_dot, false, a_dot, (short)0, c_dot, false, false);

        // (2) Denominators as rank-1 outer product with EPS folded into C:
        //     D[M][N] = n_M * n_N + EPS
        v2f a_n;
        a_n.x = lo ? n : 0.0f;
        a_n.y = 0.0f;
        v8f c_den = {MBP_EPS, MBP_EPS, MBP_EPS, MBP_EPS,
                     MBP_EPS, MBP_EPS, MBP_EPS, MBP_EPS};
        c_den = __builtin_amdgcn_wmma_f32_16x16x4_f32(
                    false, a_n, false, a_n, (short)0, c_den, false, false);

        // (3) Pair-table BYTE offset as rank-2 form:
        //     D[M][N] = 24*e_M + 8*e_N = sizeof(float2) * (3*e_M + e_N)  (exact in f32)
        //     A[M][:] = (24e_M, 8, 0, 0),  B[:][N] = (1, e_N, 0, 0)^T
        v2f a_cmb, b_cmb;
        a_cmb.x = lo ? 24.0f * ef : 0.0f;
        a_cmb.y = lo ? 8.0f : 0.0f;
        b_cmb.x = lo ? 1.0f : 0.0f;
        b_cmb.y = lo ? ef : 0.0f;
        v8f c_cmb = {};
        c_cmb = __builtin_amdgcn_wmma_f32_16x16x4_f32(
                    false, a_cmb, false, b_cmb, (short)0, c_cmb, false, false);

        float asum = 0.0f;
        #pragma unroll
        for (int v = 0; v < 8; ++v) {
            int   M     = v + mbase;
            float denom = c_den[v];
            // fast reciprocal + 1 Newton step (denom in [1e-12, ~4e3], always normal)
            float r0   = __builtin_amdgcn_rcpf(denom);
            float rr   = fmaf(fmaf(-denom, r0, 1.0f), r0, r0);
            float cosT = c_dot[v] * rr;
            int   cb   = (int)c_cmb[v];              // byte offset, v_cvt_i32_f32 exact
            float2 kc  = *(const float2*)(tb + cb);  // (k_angle, cos0) via LDS, no shift
            float  d   = cosT - kc.y;
            float  e   = kc.x * d * d;
            asum += (M == row) ? 0.0f : e;           // exclude j == k diagonal
        }
        dsum += (double)asum;
    }

    // Wave32 reduction in double, then one f64 atomic per wave.
    #pragma unroll
    for (int off = 16; off > 0; off >>= 1)
        dsum += __shfl_down(dsum, off);
    if (lane == 0)
        atomicAdd(acc, dsum);
}

__global__ void mbp_finalize(const double* __restrict__ acc, float* __restrict__ out) {
    out[0] = (float)acc[0];
}

extern "C" void kernel_launch(void* const* d_in, const int* in_sizes, int n_in,
                              void* d_out, int out_size, void* d_ws, size_t ws_size,
                              hipStream_t stream) {
    const float* pos     = (const float*)d_in[0];
    const int*   edge    = (const int*)  d_in[1]; // [2, n_edges] flat
    const int*   elem    = (const int*)  d_in[2];
    const int*   plt     = (const int*)  d_in[3];
    const float* k_angle = (const float*)d_in[4];
    const float* cos0    = (const float*)d_in[5];

    const int num_atoms = in_sizes[2];
    const int n_edges   = in_sizes[1] / 2;
    const int* edge_tgt = edge + n_edges;        // second row = targets
    double* acc = (double*)d_ws;

    mbp_zero_acc<<<1, 1, 0, stream>>>(acc);

    const int waves_per_block = 8;               // 256 threads = 8 wave32
    int blocks = (num_atoms + waves_per_block - 1) / waves_per_block;
    if (blocks > 16384) blocks = 16384;

    mbp_kernel<<<blocks, 256, 0, stream>>>(pos, edge_tgt, elem, k_angle, cos0, plt,
                                           num_atoms, acc);

    mbp_finalize<<<1, 1, 0, stream>>>(acc, (float*)d_out);
}